// BilevelInvariantPointGraphAttention_37134287242024
// MI455X (gfx1250) — compile-verified
//
#include <hip/hip_runtime.h>

#define C_S    128
#define C_Z    64
#define CA     16
#define HH     4
#define NRES   1000
#define ERES   4000
#define NATOM  14000
#define EATOM  784000

typedef __attribute__((ext_vector_type(2))) float v2f;
typedef __attribute__((ext_vector_type(8))) float v8f;

__device__ __forceinline__ v8f wmma4(v2f a, v2f b, v8f c) {
  // D = A(16x4,f32) * B(4x16,f32) + C ; chained over K in steps of 4
  return __builtin_amdgcn_wmma_f32_16x16x4_f32(false, a, false, b, (short)0, c,
                                               false, false);
}

// monotone float<->uint encoding for atomic segment-max
__device__ __forceinline__ unsigned fenc(float f) {
  unsigned u = __float_as_uint(f);
  return (u & 0x80000000u) ? ~u : (u | 0x80000000u);
}
__device__ __forceinline__ float fdec(unsigned u) {
  return __uint_as_float((u & 0x80000000u) ? (u & 0x7FFFFFFFu) : ~u);
}

// ---------------- K1: atom projections Q = A*wq+bq, KV = A*wkv+bkv ----------
__global__ void __launch_bounds__(32)
proj_kernel(const float* __restrict__ afeat,
            const float* __restrict__ wq, const float* __restrict__ bq,
            const float* __restrict__ wkv, const float* __restrict__ bkv,
            float* __restrict__ Q, float* __restrict__ KV) {
  const int tile = blockIdx.x;
  const int lane = threadIdx.x;
  const int row = lane & 15;
  const int hi  = lane >> 4;
  const int arow = tile * 16 + row;

  v2f afrag[4];
#pragma unroll
  for (int kc = 0; kc < 4; ++kc) {
    int k0 = kc * 4 + hi * 2;                      // A-frag: lane<16 -> K=k0,k0+1 ; lane>=16 -> K=k0+2,k0+3
    afrag[kc].x = afeat[arow * CA + k0];
    afrag[kc].y = afeat[arow * CA + k0 + 1];
  }
#pragma unroll
  for (int t = 0; t < 4; ++t) {                    // Q: 4 column tiles of 16
    const int n0 = t * 16;
    v8f c;
    float bias = bq[n0 + row];
#pragma unroll
    for (int r = 0; r < 8; ++r) c[r] = bias;
#pragma unroll
    for (int kc = 0; kc < 4; ++kc) {
      int k0 = kc * 4 + hi * 2;
      v2f b;
      b.x = wq[(k0 + 0) * 64 + n0 + row];
      b.y = wq[(k0 + 1) * 64 + n0 + row];
      c = wmma4(afrag[kc], b, c);
    }
#pragma unroll
    for (int r = 0; r < 8; ++r)
      Q[(tile * 16 + r + 8 * hi) * 64 + n0 + row] = c[r];
  }
#pragma unroll
  for (int t = 0; t < 8; ++t) {                    // KV: 8 column tiles of 16
    const int n0 = t * 16;
    v8f c;
    float bias = bkv[n0 + row];
#pragma unroll
    for (int r = 0; r < 8; ++r) c[r] = bias;
#pragma unroll
    for (int kc = 0; kc < 4; ++kc) {
      int k0 = kc * 4 + hi * 2;
      v2f b;
      b.x = wkv[(k0 + 0) * 128 + n0 + row];
      b.y = wkv[(k0 + 1) * 128 + n0 + row];
      c = wmma4(afrag[kc], b, c);
    }
#pragma unroll
    for (int r = 0; r < 8; ++r)
      KV[(tile * 16 + r + 8 * hi) * 128 + n0 + row] = c[r];
  }
}

// ---------------- K2: res_level_bias = relu(cat@w1+b1)@w2+b2 ----------------
__device__ __forceinline__ float cat_feat(const float* ef, const float* node,
                                          int er, int i0, int i1, int c) {
  if (c < C_Z) return ef[er * C_Z + c];
  if (c < C_Z + C_S) return node[i0 * C_S + (c - C_Z)];
  return node[i1 * C_S + (c - C_Z - C_S)];
}

__global__ void __launch_bounds__(32)
res_bias_kernel(const float* __restrict__ node, const float* __restrict__ ef,
                const int* __restrict__ reidx,
                const float* __restrict__ w1, const float* __restrict__ b1,
                const float* __restrict__ w2, const float* __restrict__ b2,
                float* __restrict__ resb) {
  __shared__ float h1[16][17];
  const int tile = blockIdx.x;
  const int lane = threadIdx.x;
  const int row = lane & 15;
  const int hi  = lane >> 4;
  const int er  = tile * 16 + row;
  const int i0  = reidx[er];
  const int i1  = reidx[ERES + er];

  v8f c;
  float bias = b1[row];
#pragma unroll
  for (int r = 0; r < 8; ++r) c[r] = bias;
  for (int kc = 0; kc < 80; ++kc) {                // K = 320 = 80 * 4
    int k0 = kc * 4 + hi * 2;
    v2f a, b;
    a.x = cat_feat(ef, node, er, i0, i1, k0);
    a.y = cat_feat(ef, node, er, i0, i1, k0 + 1);
    b.x = w1[(k0 + 0) * CA + row];
    b.y = w1[(k0 + 1) * CA + row];
    c = wmma4(a, b, c);
  }
#pragma unroll
  for (int r = 0; r < 8; ++r) h1[r + 8 * hi][row] = fmaxf(c[r], 0.0f);
  __syncthreads();
#pragma unroll
  for (int j = 0; j < 2; ++j) {                    // 16x16 @ 16x4, scalar
    int o = hi * 2 + j;
    float acc = b2[o];
#pragma unroll
    for (int k = 0; k < 16; ++k) acc += h1[row][k] * w2[k * HH + o];
    resb[er * HH + o] = acc;
  }
}

// ---------------- K3: per-edge RBF MLP -> atom_dist_bias -------------------
__global__ void __launch_bounds__(32)
dist_bias_kernel(const float* __restrict__ coords, const int* __restrict__ ceidx,
                 const float* __restrict__ w1, const float* __restrict__ b1,
                 const float* __restrict__ w2, const float* __restrict__ b2,
                 const float* __restrict__ w3, const float* __restrict__ b3,
                 float* __restrict__ distb) {
  __shared__ float dsh[16];
  __shared__ float hsh[16][17];
  const int tile = blockIdx.x;
  const int lane = threadIdx.x;
  const int row = lane & 15;
  const int hi  = lane >> 4;
  const int e   = tile * 16 + row;

  if (lane < 16) {
    int dst = ceidx[e];
    int src = ceidx[EATOM + e];
    float dx = coords[dst * 3 + 0] - coords[src * 3 + 0] + 1e-8f;
    float dy = coords[dst * 3 + 1] - coords[src * 3 + 1] + 1e-8f;
    float dz = coords[dst * 3 + 2] - coords[src * 3 + 2] + 1e-8f;
    dsh[row] = sqrtf(dx * dx + dy * dy + dz * dz);
  }
  __syncthreads();
  const float d = dsh[row];
  const float inv_sigma = 1.0f / 1.25f;            // sigma = 20/16
  const float mu_step = 20.0f / 15.0f;             // linspace(0,20,16)

  v8f c1;
  {
    float bias = b1[row];
#pragma unroll
    for (int r = 0; r < 8; ++r) c1[r] = bias;
  }
#pragma unroll
  for (int kc = 0; kc < 4; ++kc) {
    int k0 = kc * 4 + hi * 2;
    v2f a, b;
    float t0 = (d - mu_step * (float)(k0 + 0)) * inv_sigma;
    float t1 = (d - mu_step * (float)(k0 + 1)) * inv_sigma;
    a.x = __expf(-t0 * t0);
    a.y = __expf(-t1 * t1);
    b.x = w1[(k0 + 0) * 16 + row];
    b.y = w1[(k0 + 1) * 16 + row];
    c1 = wmma4(a, b, c1);
  }
#pragma unroll
  for (int r = 0; r < 8; ++r) hsh[r + 8 * hi][row] = fmaxf(c1[r], 0.0f);
  __syncthreads();

  v8f c2;
  {
    float bias = b2[row];
#pragma unroll
    for (int r = 0; r < 8; ++r) c2[r] = bias;
  }
#pragma unroll
  for (int kc = 0; kc < 4; ++kc) {
    int k0 = kc * 4 + hi * 2;
    v2f a, b;
    a.x = hsh[row][k0];
    a.y = hsh[row][k0 + 1];
    b.x = w2[(k0 + 0) * 16 + row];
    b.y = w2[(k0 + 1) * 16 + row];
    c2 = wmma4(a, b, c2);
  }
  __syncthreads();
#pragma unroll
  for (int r = 0; r < 8; ++r) hsh[r + 8 * hi][row] = fmaxf(c2[r], 0.0f);
  __syncthreads();
#pragma unroll
  for (int j = 0; j < 2; ++j) {                    // 16x16 @ 16x4 scalar
    int o = hi * 2 + j;
    float acc = b3[o];
#pragma unroll
    for (int k = 0; k < 16; ++k) acc += hsh[row][k] * w3[k * HH + o];
    distb[e * HH + o] = acc;
  }
}

// ---------------- K4: R = qk/4 + biases ; segment max / block sums ---------
__global__ void qk_kernel(const int* __restrict__ ceidx, const int* __restrict__ a2r,
                          const float* __restrict__ Q, const float* __restrict__ KV,
                          const float* __restrict__ resb, const float* __restrict__ distb,
                          float* __restrict__ R, float* __restrict__ block_r,
                          float* __restrict__ cnt, unsigned* __restrict__ m_atom) {
  int t = blockIdx.x * blockDim.x + threadIdx.x;
  if (t >= EATOM * HH) return;
  int e = t >> 2, h = t & 3;
  int dst = ceidx[e];
  int src = ceidx[EATOM + e];
  int re  = a2r[e];
  const float* q = Q + src * 64 + h * 16;
  const float* k = KV + dst * 128 + h * 32;
  float acc = 0.0f;
#pragma unroll
  for (int c = 0; c < 16; ++c) acc += q[c] * k[c];
  float r = acc * 0.25f + resb[re * HH + h] + distb[t];
  R[t] = r;
  atomicMax(&m_atom[src * HH + h], fenc(r));
  atomicAdd(&block_r[re * HH + h], r);
  if (h == 0) atomicAdd(&cnt[re], 1.0f);
}

// ---------------- K5a: block_r /= clip(cnt,1) ; beta seg-max ---------------
__global__ void block_kernel(const float* __restrict__ cnt, float* __restrict__ block_r,
                             const int* __restrict__ reidx, unsigned* __restrict__ beta_m) {
  int t = blockIdx.x * blockDim.x + threadIdx.x;
  if (t >= ERES * HH) return;
  int re = t >> 2, h = t & 3;
  float v = block_r[t] / fmaxf(cnt[re], 1.0f);
  block_r[t] = v;
  int rd = reidx[ERES + re];
  atomicMax(&beta_m[rd * HH + h], fenc(v));
}

// ---------------- K5b: alpha exp + segment sum -----------------------------
__global__ void exp_kernel(const int* __restrict__ ceidx, float* __restrict__ R,
                           const unsigned* __restrict__ m_atom, float* __restrict__ s_atom) {
  int t = blockIdx.x * blockDim.x + threadIdx.x;
  if (t >= EATOM * HH) return;
  int e = t >> 2, h = t & 3;
  int src = ceidx[EATOM + e];
  float ev = __expf(R[t] - fdec(m_atom[src * HH + h]));
  R[t] = ev;
  atomicAdd(&s_atom[src * HH + h], ev);
}

// ---------------- K5c / K6: beta exp + sum, then normalize -----------------
__global__ void beta_exp_kernel(const float* __restrict__ block_r,
                                const int* __restrict__ reidx,
                                const unsigned* __restrict__ beta_m,
                                float* __restrict__ beta_s, float* __restrict__ beta) {
  int t = blockIdx.x * blockDim.x + threadIdx.x;
  if (t >= ERES * HH) return;
  int re = t >> 2, h = t & 3;
  int rd = reidx[ERES + re];
  float ev = __expf(block_r[t] - fdec(beta_m[rd * HH + h]));
  beta[t] = ev;
  atomicAdd(&beta_s[rd * HH + h], ev);
}

__global__ void beta_fin_kernel(const int* __restrict__ reidx,
                                const float* __restrict__ beta_s, float* __restrict__ beta) {
  int t = blockIdx.x * blockDim.x + threadIdx.x;
  if (t >= ERES * HH) return;
  int re = t >> 2, h = t & 3;
  int rd = reidx[ERES + re];
  beta[t] = beta[t] / (beta_s[rd * HH + h] + 1e-16f);
}

// ---------------- K7: w*v scatter into attn64 ------------------------------
__global__ void accum_kernel(const int* __restrict__ ceidx, const int* __restrict__ a2r,
                             const float* __restrict__ R, const float* __restrict__ s_atom,
                             const float* __restrict__ beta, const float* __restrict__ KV,
                             float* __restrict__ attn64) {
  int t = blockIdx.x * blockDim.x + threadIdx.x;
  if (t >= EATOM * HH) return;
  int e = t >> 2, h = t & 3;
  int dst = ceidx[e];
  int src = ceidx[EATOM + e];
  int re  = a2r[e];
  float alpha = R[t] / (s_atom[src * HH + h] + 1e-16f);
  float w = alpha * beta[re * HH + h];
  const float* v = KV + dst * 128 + h * 32 + 16;
  float* o = attn64 + src * 64 + h * 16;
#pragma unroll
  for (int c = 0; c < 16; ++c) atomicAdd(&o[c], w * v[c]);
}

// ---------------- K8: out = attn64 @ wo + bo (WMMA) ------------------------
__global__ void __launch_bounds__(32)
out_kernel(const float* __restrict__ attn64, const float* __restrict__ wo,
           const float* __restrict__ bo, float* __restrict__ out) {
  const int tile = blockIdx.x;
  const int lane = threadIdx.x;
  const int row = lane & 15;
  const int hi  = lane >> 4;
  const int arow = tile * 16 + row;
  v8f c;
  float bias = bo[row];
#pragma unroll
  for (int r = 0; r < 8; ++r) c[r] = bias;
#pragma unroll
  for (int kc = 0; kc < 16; ++kc) {                // K = 64
    int k0 = kc * 4 + hi * 2;
    v2f a, b;
    a.x = attn64[arow * 64 + k0];
    a.y = attn64[arow * 64 + k0 + 1];
    b.x = wo[(k0 + 0) * CA + row];
    b.y = wo[(k0 + 1) * CA + row];
    c = wmma4(a, b, c);
  }
#pragma unroll
  for (int r = 0; r < 8; ++r)
    out[(tile * 16 + r + 8 * hi) * CA + row] = c[r];
}

// ---------------------------------------------------------------------------
extern "C" void kernel_launch(void* const* d_in, const int* in_sizes, int n_in,
                              void* d_out, int out_size, void* d_ws, size_t ws_size,
                              hipStream_t stream) {
  (void)in_sizes; (void)n_in; (void)out_size; (void)ws_size;
  const float* node   = (const float*)d_in[0];
  const float* ef     = (const float*)d_in[1];
  const int*   reidx  = (const int*)d_in[2];
  const float* afeat  = (const float*)d_in[3];
  const float* coords = (const float*)d_in[4];
  const int*   ceidx  = (const int*)d_in[5];
  const int*   a2r    = (const int*)d_in[6];
  const float* rbf_w1 = (const float*)d_in[7];
  const float* rbf_b1 = (const float*)d_in[8];
  const float* rbf_w2 = (const float*)d_in[9];
  const float* rbf_b2 = (const float*)d_in[10];
  const float* rbf_w3 = (const float*)d_in[11];
  const float* rbf_b3 = (const float*)d_in[12];
  const float* rb_w1  = (const float*)d_in[13];
  const float* rb_b1  = (const float*)d_in[14];
  const float* rb_w2  = (const float*)d_in[15];
  const float* rb_b2  = (const float*)d_in[16];
  const float* wq     = (const float*)d_in[17];
  const float* bq     = (const float*)d_in[18];
  const float* wkv    = (const float*)d_in[19];
  const float* bkv    = (const float*)d_in[20];
  const float* wo     = (const float*)d_in[21];
  const float* bo     = (const float*)d_in[22];
  float* out = (float*)d_out;

  // workspace layout (f32 elements)
  float* ws      = (float*)d_ws;
  float* Q       = ws;                                   // 896000
  float* KV      = Q + (size_t)NATOM * 64;               // 1792000
  float* distb   = KV + (size_t)NATOM * 128;             // 3136000
  float* resb    = distb + (size_t)EATOM * HH;           // 16000
  float* R       = resb + (size_t)ERES * HH;             // 3136000
  float* beta    = R + (size_t)EATOM * HH;               // 16000
  // ---- zero-init region starts here ----
  unsigned* m_atom = (unsigned*)(beta + (size_t)ERES * HH);  // 56000 (uint enc: 0 == -inf)
  float* s_atom    = (float*)(m_atom + (size_t)NATOM * HH);  // 56000
  float* block_r   = s_atom + (size_t)NATOM * HH;            // 16000
  float* cnt       = block_r + (size_t)ERES * HH;            // 4000
  unsigned* beta_m = (unsigned*)(cnt + ERES);                // 4000
  float* beta_s    = (float*)(beta_m + (size_t)NRES * HH);   // 4000
  float* attn64    = beta_s + (size_t)NRES * HH;             // 896000
  size_t zero_bytes =
      ((size_t)NATOM * HH * 2 + (size_t)ERES * HH + ERES +
       (size_t)NRES * HH * 2 + (size_t)NATOM * 64) * sizeof(float);
  hipMemsetAsync(m_atom, 0, zero_bytes, stream);

  const int EH  = EATOM * HH;                // 3,136,000
  const int REH = ERES * HH;                 // 16,000

  proj_kernel<<<NATOM / 16, 32, 0, stream>>>(afeat, wq, bq, wkv, bkv, Q, KV);
  res_bias_kernel<<<ERES / 16, 32, 0, stream>>>(node, ef, reidx, rb_w1, rb_b1,
                                                rb_w2, rb_b2, resb);
  dist_bias_kernel<<<EATOM / 16, 32, 0, stream>>>(coords, ceidx, rbf_w1, rbf_b1,
                                                  rbf_w2, rbf_b2, rbf_w3, rbf_b3,
                                                  distb);
  qk_kernel<<<(EH + 255) / 256, 256, 0, stream>>>(ceidx, a2r, Q, KV, resb, distb,
                                                  R, block_r, cnt, m_atom);
  block_kernel<<<(REH + 255) / 256, 256, 0, stream>>>(cnt, block_r, reidx, beta_m);
  exp_kernel<<<(EH + 255) / 256, 256, 0, stream>>>(ceidx, R, m_atom, s_atom);
  beta_exp_kernel<<<(REH + 255) / 256, 256, 0, stream>>>(block_r, reidx, beta_m,
                                                         beta_s, beta);
  beta_fin_kernel<<<(REH + 255) / 256, 256, 0, stream>>>(reidx, beta_s, beta);
  accum_kernel<<<(EH + 255) / 256, 256, 0, stream>>>(ceidx, a2r, R, s_atom, beta,
                                                     KV, attn64);
  out_kernel<<<NATOM / 16, 32, 0, stream>>>(attn64, wo, bo, out);
}